// BatchAll_76639396429874
// MI455X (gfx1250) — compile-verified
//
#include <hip/hip_runtime.h>
#include <math.h>

#define N_    4096
#define D_    128
#define K_    4
#define MARGIN_ 0.02f
#define EPS_  1e-12f

typedef __attribute__((ext_vector_type(2))) float v2f;
typedef __attribute__((ext_vector_type(4))) float v4f;
typedef __attribute__((ext_vector_type(8))) float v8f;

// ---------------- Kernel 1: row squared norms -------------------------------
__global__ void __launch_bounds__(256) sq_kernel(const float* __restrict__ X,
                                                 float* __restrict__ sq) {
    int i = blockIdx.x * 256 + threadIdx.x;
    if (i >= N_) return;
    const v4f* row = (const v4f*)(X + i * D_);
    float s = 0.f;
#pragma unroll 8
    for (int q = 0; q < D_ / 4; ++q) {
        v4f v = row[q];
        s += v.x * v.x + v.y * v.y + v.z * v.z + v.w * v.w;
    }
    sq[i] = s;
}

// ---------------- Kernel 2: positive distances (4096 x 3) -------------------
__global__ void __launch_bounds__(256) posdist_kernel(const float* __restrict__ X,
                                                      const float* __restrict__ sq,
                                                      float* __restrict__ posD) {
    int t = blockIdx.x * 256 + threadIdx.x;
    if (t >= N_ * (K_ - 1)) return;
    int i = t / (K_ - 1);
    int o = t - i * (K_ - 1);
    int pic = i & (K_ - 1);
    int cs  = i & ~(K_ - 1);
    int j = cs + o + (o >= pic ? 1 : 0);
    const v4f* xi = (const v4f*)(X + i * D_);
    const v4f* xj = (const v4f*)(X + j * D_);
    float dot = 0.f;
#pragma unroll 8
    for (int q = 0; q < D_ / 4; ++q) {
        v4f a = xi[q], b = xj[q];
        dot += a.x * b.x + a.y * b.y + a.z * b.z + a.w * b.w;
    }
    float d2 = sq[i] + sq[j] - 2.f * dot;
    posD[t] = sqrtf(fmaxf(d2, EPS_));
}

// ---------------- Kernel 3: fused Gram (WMMA f32) + triplet accumulate ------
// grid.x = 256 row tiles (16 rows each), grid.y = 8 column groups.
// 8 waves per block; wave w handles column tiles (blockIdx.y*8 + w)*4 + t, t=0..3.
__global__ void __launch_bounds__(256) triplet_gram_kernel(
    const float* __restrict__ X, const float* __restrict__ sq,
    const float* __restrict__ posD, float* __restrict__ accF,
    unsigned* __restrict__ accU, unsigned* __restrict__ anchorCnt) {

    const int lane = threadIdx.x & 31;
    const int wave = threadIdx.x >> 5;
    const int lrow = lane & 15;          // row/col within tile this lane loads
    const int half = lane >> 4;          // 0: K 0,1 | 1: K 2,3 (A/B layout)
    const int rowBase = blockIdx.x * 16;

    // Per-lane anchor metadata: C-matrix element r maps to M = r + 8*half.
    float sqi[8];
    float pd[8][3];
#pragma unroll
    for (int r = 0; r < 8; ++r) {
        int i = rowBase + r + 8 * half;
        sqi[r] = sq[i];
        pd[r][0] = posD[i * 3 + 0];
        pd[r][1] = posD[i * 3 + 1];
        pd[r][2] = posD[i * 3 + 2];
    }

    const float* arow = X + (rowBase + lrow) * D_ + 2 * half;

    float tripSum = 0.f, negSum = 0.f;
    unsigned valid = 0;
    unsigned cnt[8] = {0, 0, 0, 0, 0, 0, 0, 0};

    for (int t = 0; t < 4; ++t) {
        const int colTile = (blockIdx.y * 8 + wave) * 4 + t;
        const int colBase = colTile * 16;
        const float* brow = X + (colBase + lrow) * D_ + 2 * half;

        v8f c = {0.f, 0.f, 0.f, 0.f, 0.f, 0.f, 0.f, 0.f};
#pragma unroll
        for (int kk = 0; kk < D_; kk += 4) {
            v2f a = *(const v2f*)(arow + kk);
            v2f b = *(const v2f*)(brow + kk);
            // D = A(16x4) * B(4x16) + C  — v_wmma_f32_16x16x4_f32
            c = __builtin_amdgcn_wmma_f32_16x16x4_f32(
                /*neg_a=*/false, a, /*neg_b=*/false, b,
                /*c_mod=*/(short)0, c, /*reuse_a=*/false, /*reuse_b=*/false);
        }

        const int j = colBase + lrow;          // column index for this lane
        const float sqj = sq[j];
        const int jc = j >> 2;                 // class of column j (k=4)
#pragma unroll
        for (int r = 0; r < 8; ++r) {
            const int i = rowBase + r + 8 * half;
            if ((i >> 2) == jc) continue;      // same class: not a negative
            float d2 = sqi[r] + sqj - 2.f * c[r];
            float d = sqrtf(fmaxf(d2, EPS_));
            negSum += d;
#pragma unroll
            for (int p = 0; p < 3; ++p) {
                float tr = pd[r][p] - d + MARGIN_;
                if (tr > 0.f) { tripSum += tr; ++valid; ++cnt[r]; }
            }
        }
    }

    // Wave-wide reduce of the scalar accumulators (wave32).
#pragma unroll
    for (int m = 16; m >= 1; m >>= 1) {
        tripSum += __shfl_xor(tripSum, m, 32);
        negSum  += __shfl_xor(negSum,  m, 32);
        valid   += __shfl_xor(valid,   m, 32);
    }
    if (lane == 0) {
        atomicAdd(&accF[0], tripSum);
        atomicAdd(&accF[1], negSum);
        atomicAdd(&accU[0], valid);
    }

    // Per-anchor valid counts: reduce within each 16-lane half (same anchors).
#pragma unroll
    for (int r = 0; r < 8; ++r) {
        unsigned c8 = cnt[r];
#pragma unroll
        for (int m = 8; m >= 1; m >>= 1) c8 += __shfl_xor(c8, m, 32);
        if (lrow == 0 && c8 != 0u)
            atomicAdd(&anchorCnt[rowBase + r + 8 * half], c8);
    }
}

// ---------------- Kernel 4: finalize the 4 scalars --------------------------
__global__ void __launch_bounds__(256) finalize_kernel(
    const float* __restrict__ posD, const float* __restrict__ accF,
    const unsigned* __restrict__ accU, const unsigned* __restrict__ anchorCnt,
    float* __restrict__ out) {
    __shared__ float    sf[256];
    __shared__ unsigned su[256];
    int tid = threadIdx.x;

    float ps = 0.f;
    for (int idx = tid; idx < N_ * (K_ - 1); idx += 256) ps += posD[idx];
    unsigned zc = 0;
    for (int idx = tid; idx < N_; idx += 256) zc += (anchorCnt[idx] == 0u) ? 1u : 0u;

    sf[tid] = ps;
    su[tid] = zc;
    __syncthreads();
    for (int s = 128; s > 0; s >>= 1) {
        if (tid < s) { sf[tid] += sf[tid + s]; su[tid] += su[tid + s]; }
        __syncthreads();
    }
    if (tid == 0) {
        float trip = accF[0];
        float neg  = accF[1];
        unsigned valid = accU[0];
        out[0] = (valid > 0u) ? trip / (float)valid : 0.f;          // loss
        out[1] = (float)su[0] / (float)N_;                          // accuracy
        out[2] = sf[0] / (float)(N_ * (K_ - 1));                    // mean pos dist
        out[3] = neg / ((float)N_ * (float)(N_ - K_));              // mean neg dist
    }
}

// ---------------- Host-side launch ------------------------------------------
extern "C" void kernel_launch(void* const* d_in, const int* in_sizes, int n_in,
                              void* d_out, int out_size, void* d_ws, size_t ws_size,
                              hipStream_t stream) {
    const float* X = (const float*)d_in[0];
    (void)in_sizes; (void)n_in; (void)out_size; (void)ws_size;

    // Workspace layout (floats):
    //  [0, 4096)              sq
    //  [4096, 16384)          posD (4096 x 3)
    //  [16384, 16386)         accF: tripSum, negSum
    //  [16386, 16387)         accU: valid count (as uint)
    //  [16387, 16387+4096)    anchorCnt (uint)
    float* ws        = (float*)d_ws;
    float* sq        = ws;
    float* posD      = ws + N_;
    float* accF      = ws + N_ + N_ * (K_ - 1);
    unsigned* accU   = (unsigned*)(accF + 2);
    unsigned* aCnt   = accU + 1;
    float* out       = (float*)d_out;

    // Zero accumulators + per-anchor counts (graph-capturable memset).
    hipMemsetAsync(accF, 0, (3 + N_) * sizeof(float), stream);

    sq_kernel<<<N_ / 256, 256, 0, stream>>>(X, sq);
    posdist_kernel<<<(N_ * (K_ - 1) + 255) / 256, 256, 0, stream>>>(X, sq, posD);
    triplet_gram_kernel<<<dim3(N_ / 16, 8), 256, 0, stream>>>(X, sq, posD, accF, accU, aCnt);
    finalize_kernel<<<1, 256, 0, stream>>>(posD, accF, accU, aCnt, out);
}